// RAFF_8555574853896
// MI455X (gfx1250) — compile-verified
//
#include <hip/hip_runtime.h>
#include <hip/hip_bf16.h>

typedef float v2f __attribute__((ext_vector_type(2)));
typedef float v8f __attribute__((ext_vector_type(8)));
typedef int ai4 __attribute__((vector_size(16)));   // matches builtin's V4i

constexpr int B_ = 2048, P_ = 16, S_ = 32, D_ = 1024, F_ = 512, H_ = 512;

#if __has_builtin(__builtin_amdgcn_global_load_async_to_lds_b128) && \
    __has_builtin(__builtin_amdgcn_s_wait_asynccnt)
#define HAVE_ASYNC_LDS 1
#else
#define HAVE_ASYNC_LDS 0
#endif

// 16-byte global -> LDS copy. Async (ASYNCcnt-tracked, no VGPR round trip) when
// the gfx1250 builtins are available; both addresses must be 16B aligned.
__device__ __forceinline__ void async_cp16(const float* g, float* l) {
#if HAVE_ASYNC_LDS
    __builtin_amdgcn_global_load_async_to_lds_b128(
        (ai4 __attribute__((address_space(1)))*)(g),
        (ai4 __attribute__((address_space(3)))*)(l), 0, 0);
#else
    *(float4*)l = *(const float4*)g;
#endif
}

template <int N>
__device__ __forceinline__ void wait_async() {
#if HAVE_ASYNC_LDS
    __builtin_amdgcn_s_wait_asynccnt(N);
#endif
}

// D = A(16x4) * B(4x16) + C, exact f32 on the WMMA pipe.
// A: lane(0..15)=M, VGPR pair = K{0,1}; lanes 16..31 = K{2,3}. B mirrored with lane=N.
// C/D: VGPR v -> M = v + 8*(lane>=16), N = lane&15.
__device__ __forceinline__ v8f wmma_f32(v2f a, v2f b, v8f c) {
    return __builtin_amdgcn_wmma_f32_16x16x4_f32(false, a, false, b, (short)0, c, false, false);
}

// ---------------------------------------------------------------------------
// Generic GEMM: C[m,n] = sum_k A[m*K+k]*W[n*K+k] + bias[n]
// 256 threads = 8 waves; workgroup tile 32(M) x 64(N); K chunk 32;
// double-buffered LDS fed by async global->LDS copies.
// M % 32 == 0, N % 64 == 0, K % 32 == 0 (true for all uses here).
// ---------------------------------------------------------------------------
__global__ __launch_bounds__(256) void gemm_nt(const float* __restrict__ A,
                                               const float* __restrict__ W,
                                               const float* __restrict__ bias,
                                               float* __restrict__ C,
                                               int M, int N, int K) {
    constexpr int KC = 32, LDK = KC + 4;   // 36-float rows: 16B aligned, conflict-free
    __shared__ float sA[2][32][LDK];
    __shared__ float sB[2][64][LDK];
    const int tid  = threadIdx.x;
    const int wave = tid >> 5;
    const int lane = tid & 31;
    const int m0 = blockIdx.x * 32;
    const int n0 = blockIdx.y * 64;
    const int wm = (wave >> 2) * 16;   // 0 or 16
    const int wn = (wave & 3) * 16;    // 0..48
    const int fr = lane & 15;
    const int kh = (lane >> 4) * 2;    // 0 or 2
    const int ra = tid >> 3, ca = (tid & 7) * 4;          // A slot: 32x32, 16B/thread
    const int rb1 = ra + 32;                              // B slot rows 32..63
    v8f acc = {};

    auto issue = [&](int kb, int bf) {
        async_cp16(A + (size_t)(m0 + ra) * K + kb + ca, &sA[bf][ra][ca]);
        async_cp16(W + (size_t)(n0 + ra) * K + kb + ca, &sB[bf][ra][ca]);
        async_cp16(W + (size_t)(n0 + rb1) * K + kb + ca, &sB[bf][rb1][ca]);
    };

    const int nch = K / KC;
    issue(0, 0);
    for (int ch = 0; ch < nch; ++ch) {
        const int bf = ch & 1;
        if (ch + 1 < nch) {
            issue((ch + 1) * KC, bf ^ 1);
            wait_async<3>();           // retire current chunk; keep prefetch in flight
        } else {
            wait_async<0>();
        }
        __syncthreads();
#pragma unroll
        for (int ks = 0; ks < KC; ks += 4) {
            v2f a = *(const v2f*)&sA[bf][wm + fr][ks + kh];
            v2f b = *(const v2f*)&sB[bf][wn + fr][ks + kh];
            acc = wmma_f32(a, b, acc);
        }
        __syncthreads();
    }
    const int cm = m0 + wm + (lane >> 4) * 8;
    const int cn = n0 + wn + fr;
    const float bv = bias ? bias[cn] : 0.0f;
#pragma unroll
    for (int v = 0; v < 8; ++v) C[(size_t)(cm + v) * N + cn] = acc[v] + bv;
}

// ---------------------------------------------------------------------------
// Classifier: hpart[b,s,g] = sum_{n in group g(64)} relu(q[b,s,:]·w1[s,n,:] + b1[s,n]) * w2[s,n]
// grid: (B/32, H/64, S)
// ---------------------------------------------------------------------------
__global__ __launch_bounds__(256) void cls_kernel(const float* __restrict__ q,
                                                  const float* __restrict__ cls_w1,
                                                  const float* __restrict__ cls_b1,
                                                  const float* __restrict__ cls_w2,
                                                  float* __restrict__ hpart) {
    constexpr int KC = 32, LDK = KC + 4;
    __shared__ float sA[2][32][LDK];
    __shared__ float sB[2][64][LDK];
    __shared__ float red[32][4];
    const int tid = threadIdx.x, wave = tid >> 5, lane = tid & 31;
    const int m0 = blockIdx.x * 32;
    const int g  = blockIdx.y;        // 64-wide n group
    const int s  = blockIdx.z;
    const int n0 = g * 64;
    const int wm = (wave >> 2) * 16, wn = (wave & 3) * 16;
    const int fr = lane & 15, kh = (lane >> 4) * 2;
    const int ra = tid >> 3, ca = (tid & 7) * 4;
    const int rb1 = ra + 32;
    const float* W = cls_w1 + (size_t)s * H_ * D_;
    v8f acc = {};

    auto issue = [&](int kb, int bf) {
        async_cp16(q + ((size_t)(m0 + ra) * S_ + s) * D_ + kb + ca, &sA[bf][ra][ca]);
        async_cp16(W + (size_t)(n0 + ra) * D_ + kb + ca, &sB[bf][ra][ca]);
        async_cp16(W + (size_t)(n0 + rb1) * D_ + kb + ca, &sB[bf][rb1][ca]);
    };

    constexpr int nch = D_ / KC;
    issue(0, 0);
    for (int ch = 0; ch < nch; ++ch) {
        const int bf = ch & 1;
        if (ch + 1 < nch) {
            issue((ch + 1) * KC, bf ^ 1);
            wait_async<3>();
        } else {
            wait_async<0>();
        }
        __syncthreads();
#pragma unroll
        for (int ks = 0; ks < KC; ks += 4) {
            v2f a = *(const v2f*)&sA[bf][wm + fr][ks + kh];
            v2f b = *(const v2f*)&sB[bf][wn + fr][ks + kh];
            acc = wmma_f32(a, b, acc);
        }
        __syncthreads();
    }
    // epilogue: relu(h + b1) * w2, reduce over 64 columns -> 32 row partials
    const int cn = n0 + wn + fr;
    const float b1v = cls_b1[(size_t)s * H_ + cn];
    const float w2v = cls_w2[(size_t)s * H_ + cn];
    float r[8];
#pragma unroll
    for (int v = 0; v < 8; ++v) r[v] = fmaxf(acc[v] + b1v, 0.0f) * w2v;
#pragma unroll
    for (int mask = 1; mask <= 8; mask <<= 1)
#pragma unroll
        for (int v = 0; v < 8; ++v) r[v] += __shfl_xor(r[v], mask, 32);
    if (fr == 0) {
        int rbase = wm + (lane >> 4) * 8;
#pragma unroll
        for (int v = 0; v < 8; ++v) red[rbase + v][wave & 3] = r[v];
    }
    __syncthreads();
    if (tid < 32) {
        float sum = red[tid][0] + red[tid][1] + red[tid][2] + red[tid][3];
        hpart[((size_t)(m0 + tid) * S_ + s) * 8 + g] = sum;
    }
}

__global__ void cls_reduce_kernel(const float* __restrict__ hpart,
                                  const float* __restrict__ cls_b2,
                                  float* __restrict__ d) {
    int idx = blockIdx.x * blockDim.x + threadIdx.x;   // b*S + s
    if (idx >= B_ * S_) return;
    int s = idx & (S_ - 1);
    const float* hp = hpart + (size_t)idx * 8;
    float sum = 0.f;
#pragma unroll
    for (int g2 = 0; g2 < 8; ++g2) sum += hp[g2];
    d[idx] = fmaxf(sum + cls_b2[s], 0.0f);
}

__global__ void softmax_d_kernel(const float* __restrict__ d,
                                 float* __restrict__ da, float* __restrict__ dv,
                                 float* __restrict__ out_tail) {
    int b = blockIdx.x * blockDim.x + threadIdx.x;
    if (b >= B_) return;
    const float* row = d + (size_t)b * S_;
    for (int half = 0; half < 2; ++half) {
        const float* r = row + half * P_;
        float mx = -1e30f;
        for (int i = 0; i < P_; ++i) mx = fmaxf(mx, r[i]);
        float e[P_], sum = 0.f;
        for (int i = 0; i < P_; ++i) { e[i] = __expf(r[i] - mx); sum += e[i]; }
        float inv = 1.0f / sum;
        float* dst = half ? dv : da;
        float* ot = out_tail + (size_t)half * B_ * P_ + (size_t)b * P_;
        for (int i = 0; i < P_; ++i) {
            float v = e[i] * inv;
            dst[(size_t)b * P_ + i] = v;
            ot[i] = v;
        }
    }
}

// ---------------------------------------------------------------------------
// Per-batch attention: scores = (q@wq^T+bq)(q@wk^T+bk)^T / 32, softmax,
// wA[t] = sum_{s<16} d_a[s]*attn[s,t]; u_a = wA @ q_b  (same for v).
// One workgroup (8 waves) per batch. q_b staged in LDS via async copies.
// ---------------------------------------------------------------------------
__global__ __launch_bounds__(256) void attn_kernel(const float* __restrict__ q,
                                                   const float* __restrict__ in_w,
                                                   const float* __restrict__ in_b,
                                                   const float* __restrict__ da,
                                                   const float* __restrict__ dv,
                                                   float* __restrict__ U) {
    constexpr int LQ = D_ + 4;           // 1028-float rows: 16B aligned, bank stride 4
    __shared__ float qs[S_ * LQ];        // ~128.5 KB
    __shared__ float qpc[32][66];
    __shared__ float kpc[32][66];
    __shared__ float sc[32][34];
    __shared__ float wAB[64];
    const int b = blockIdx.x;
    const int tid = threadIdx.x, wave = tid >> 5, lane = tid & 31;
    const int fr = lane & 15, kh = (lane >> 4) * 2;
    const float* qb = q + (size_t)b * S_ * D_;
    for (int i = tid * 4; i < S_ * D_; i += 1024) {
        int r = i >> 10, c = i & 1023;
        async_cp16(qb + i, &qs[r * LQ + c]);
    }
    wait_async<0>();
    __syncthreads();

    const float* wq = in_w;
    const float* wk = in_w + (size_t)D_ * D_;
    const int mi = wave & 1, ti = (wave >> 1) & 1;    // score tile for waves 0-3
    v8f accS = {};
    for (int ec = 0; ec < D_; ec += 64) {
        // all 8 waves: compute qp/kp chunk [32 x 64] with bias
        {
            const float* Wm = (wave < 4) ? wq : wk;
            const float* bv = (wave < 4) ? in_b : (in_b + D_);
            float* dst = (wave < 4) ? &qpc[0][0] : &kpc[0][0];
            const int w2 = wave & 3;
            const int tm = (w2 & 1) * 16;
            const int tn0 = (w2 >> 1) * 32;
            for (int tn = tn0; tn < tn0 + 32; tn += 16) {
                v8f acc = {};
                const int e0 = ec + tn;
                const float* wrow = Wm + (size_t)(e0 + fr) * D_ + kh;
#pragma unroll 4
                for (int k = 0; k < D_; k += 4) {
                    v2f a = *(const v2f*)&qs[(tm + fr) * LQ + k + kh];
                    v2f bf = *(const v2f*)(wrow + k);
                    acc = wmma_f32(a, bf, acc);
                }
                const int cm = tm + (lane >> 4) * 8;
                const int cn = tn + fr;
                const float bb = bv[e0 + fr];
#pragma unroll
                for (int v = 0; v < 8; ++v) dst[(cm + v) * 66 + cn] = acc[v] + bb;
            }
        }
        __syncthreads();
        if (wave < 4) {   // scores += qp_c @ kp_c^T over this 64-wide chunk
#pragma unroll
            for (int k = 0; k < 64; k += 4) {
                v2f a = *(const v2f*)&qpc[mi * 16 + fr][k + kh];
                v2f bf = *(const v2f*)&kpc[ti * 16 + fr][k + kh];
                accS = wmma_f32(a, bf, accS);
            }
        }
        __syncthreads();
    }
    if (wave < 4) {
        const int cm = mi * 16 + (lane >> 4) * 8;
        const int cn = ti * 16 + fr;
#pragma unroll
        for (int v = 0; v < 8; ++v) sc[cm + v][cn] = accS[v] * (1.0f / 32.0f);
    }
    __syncthreads();
    if (tid < 32) {      // softmax over t for row s=tid
        float mx = -1e30f;
        for (int t = 0; t < 32; ++t) mx = fmaxf(mx, sc[tid][t]);
        float e[32], sum = 0.f;
        for (int t = 0; t < 32; ++t) { e[t] = __expf(sc[tid][t] - mx); sum += e[t]; }
        float inv = 1.0f / sum;
        for (int t = 0; t < 32; ++t) sc[tid][t] = e[t] * inv;
    }
    __syncthreads();
    if (tid < 32) {      // wA[t], wV[t]
        float wa = 0.f, wv = 0.f;
        for (int s2 = 0; s2 < 16; ++s2) wa += da[(size_t)b * P_ + s2] * sc[s2][tid];
        for (int s2 = 0; s2 < 16; ++s2) wv += dv[(size_t)b * P_ + s2] * sc[16 + s2][tid];
        wAB[tid] = wa; wAB[32 + tid] = wv;
    }
    __syncthreads();
    for (int dc = tid; dc < D_; dc += 256) {   // u = w @ q_b
        float ua = 0.f, uv = 0.f;
        for (int t = 0; t < 32; ++t) {
            float qv = qs[t * LQ + dc];
            ua += wAB[t] * qv;
            uv += wAB[32 + t] * qv;
        }
        U[(size_t)b * D_ + dc] = ua;
        U[((size_t)(B_ + b)) * D_ + dc] = uv;
    }
}

__global__ void fuse_kernel(const float* __restrict__ AV, float* __restrict__ fus) {
    size_t i = ((size_t)blockIdx.x * blockDim.x + threadIdx.x) * 4;
    float4 a = *(const float4*)(AV + i);
    float4 v = *(const float4*)(AV + (size_t)B_ * D_ + i);
    float4 f;
    f.x = 0.5f * (a.x + v.x); f.y = 0.5f * (a.y + v.y);
    f.z = 0.5f * (a.z + v.z); f.w = 0.5f * (a.w + v.w);
    *(float4*)(fus + i) = f;
}

// one wave per row, rows ordered: [fusion_h | fusion_a | fusion_v], F=512
__global__ __launch_bounds__(256) void ln_relu_kernel(const float* __restrict__ Y,
                                                      const float* __restrict__ g0, const float* __restrict__ be0,
                                                      const float* __restrict__ g1, const float* __restrict__ be1,
                                                      const float* __restrict__ g2, const float* __restrict__ be2,
                                                      float* __restrict__ out) {
    const int row = blockIdx.x * 8 + (threadIdx.x >> 5);
    const int lane = threadIdx.x & 31;
    const int head = row / B_;
    const float* g  = head == 0 ? g0 : (head == 1 ? g1 : g2);
    const float* be = head == 0 ? be0 : (head == 1 ? be1 : be2);
    const float* y = Y + (size_t)row * F_;
    float vals[16], sum = 0.f;
#pragma unroll
    for (int i = 0; i < 16; ++i) { vals[i] = y[lane + i * 32]; sum += vals[i]; }
#pragma unroll
    for (int m = 1; m < 32; m <<= 1) sum += __shfl_xor(sum, m, 32);
    const float mu = sum * (1.0f / F_);
    float vs = 0.f;
#pragma unroll
    for (int i = 0; i < 16; ++i) { float t = vals[i] - mu; vs += t * t; }
#pragma unroll
    for (int m = 1; m < 32; m <<= 1) vs += __shfl_xor(vs, m, 32);
    const float rstd = rsqrtf(vs * (1.0f / F_) + 1e-5f);
    float* o = out + (size_t)row * F_;
#pragma unroll
    for (int i = 0; i < 16; ++i) {
        int c = lane + i * 32;
        o[c] = fmaxf((vals[i] - mu) * rstd * g[c] + be[c], 0.0f);
    }
}

extern "C" void kernel_launch(void* const* d_in, const int* in_sizes, int n_in,
                              void* d_out, int out_size, void* d_ws, size_t ws_size,
                              hipStream_t stream) {
    (void)in_sizes; (void)n_in; (void)out_size; (void)ws_size;
    const float* q      = (const float*)d_in[0];
    const float* in_w   = (const float*)d_in[1];
    const float* in_b   = (const float*)d_in[2];
    const float* out_w  = (const float*)d_in[3];
    const float* out_b  = (const float*)d_in[4];
    const float* cls_w1 = (const float*)d_in[5];
    const float* cls_b1 = (const float*)d_in[6];
    const float* cls_w2 = (const float*)d_in[7];
    const float* cls_b2 = (const float*)d_in[8];
    const float* fus_w  = (const float*)d_in[9];
    const float* fus_b  = (const float*)d_in[10];
    const float* fus_g  = (const float*)d_in[11];
    const float* fus_be = (const float*)d_in[12];
    const float* pa_w   = (const float*)d_in[13];
    const float* pa_b   = (const float*)d_in[14];
    const float* pa_g   = (const float*)d_in[15];
    const float* pa_be  = (const float*)d_in[16];
    const float* pv_w   = (const float*)d_in[17];
    const float* pv_b   = (const float*)d_in[18];
    const float* pv_g   = (const float*)d_in[19];
    const float* pv_be  = (const float*)d_in[20];
    float* out = (float*)d_out;
    float* ws  = (float*)d_ws;

    float* w_d     = ws;                                   // [B*S]
    float* w_hpart = w_d + B_ * S_;                        // [B*S*8]
    float* w_da    = w_hpart + (size_t)B_ * S_ * 8;        // [B*P]
    float* w_dv    = w_da + B_ * P_;                       // [B*P]
    float* w_U     = w_dv + B_ * P_;                       // [2B*D]
    float* w_P     = w_U + (size_t)2 * B_ * D_;            // [2B*D]
    float* w_AV    = w_P + (size_t)2 * B_ * D_;            // [2B*D] audio|video
    float* w_fus   = w_AV + (size_t)2 * B_ * D_;           // [B*D]
    float* w_Y     = w_fus + (size_t)B_ * D_;              // [3*B*F]

    // 1) per-position classifiers -> d[b,s]
    cls_kernel<<<dim3(B_ / 32, H_ / 64, S_), 256, 0, stream>>>(q, cls_w1, cls_b1, cls_w2, w_hpart);
    cls_reduce_kernel<<<(B_ * S_) / 256, 256, 0, stream>>>(w_hpart, cls_b2, w_d);
    // 2) d_a / d_v softmax (also final outputs)
    softmax_d_kernel<<<B_ / 256, 256, 0, stream>>>(w_d, w_da, w_dv, out + (size_t)3 * B_ * F_);
    // 3) attention (scores via WMMA, softmax, pooled u-vectors)
    attn_kernel<<<B_, 256, 0, stream>>>(q, in_w, in_b, w_da, w_dv, w_U);
    // 4) P = U @ wv^T + bv   (value projection after pooling)
    gemm_nt<<<dim3(2 * B_ / 32, D_ / 64), 256, 0, stream>>>(w_U, in_w + (size_t)2 * D_ * D_,
                                                            in_b + 2 * D_, w_P, 2 * B_, D_, D_);
    // 5) [audio|video] = P @ out_w^T + out_b
    gemm_nt<<<dim3(2 * B_ / 32, D_ / 64), 256, 0, stream>>>(w_P, out_w, out_b, w_AV, 2 * B_, D_, D_);
    // 6) fusion = 0.5*(audio+video)
    fuse_kernel<<<(B_ * D_ / 4) / 256, 256, 0, stream>>>(w_AV, w_fus);
    // 7) heads
    gemm_nt<<<dim3(B_ / 32, F_ / 64), 256, 0, stream>>>(w_fus, fus_w, fus_b, w_Y, B_, F_, D_);
    gemm_nt<<<dim3(B_ / 32, F_ / 64), 256, 0, stream>>>(w_AV, pa_w, pa_b, w_Y + (size_t)B_ * F_, B_, F_, D_);
    gemm_nt<<<dim3(B_ / 32, F_ / 64), 256, 0, stream>>>(w_AV + (size_t)B_ * D_, pv_w, pv_b,
                                                        w_Y + (size_t)2 * B_ * F_, B_, F_, D_);
    // 8) LayerNorm + ReLU -> d_out
    ln_relu_kernel<<<(3 * B_) / 8, 256, 0, stream>>>(w_Y, fus_g, fus_be, pa_g, pa_be, pv_g, pv_be, out);
}